// Mobile2Former_66254165508500
// MI455X (gfx1250) — compile-verified
//
#include <hip/hip_runtime.h>
#include <float.h>

#define HEADS 8
#define B_    32
#define C_    64
#define HW_   3136
#define M_    6
#define DIM_  192
#define INNER_ 512
#define KPH   392      // unique keys per head
#define KPH_PAD 400
#define NTILES 25

typedef float v2f __attribute__((ext_vector_type(2)));
typedef float v8f __attribute__((ext_vector_type(8)));

#if defined(__has_builtin)
#if __has_builtin(__builtin_amdgcn_s_wait_asynccnt)
#define HAVE_WAIT_ASYNC 1
#endif
#endif

// LDS layout (in floats)
#define OFF_K    0                         // 400*64   = 25600
#define OFF_Q    (KPH_PAD * C_)            // 16*64    = 1024
#define OFF_ATTN (OFF_Q + 16 * C_)         // 16*400   = 6400
#define OFF_OUT  (OFF_ATTN + 16 * KPH_PAD) // 16*64    = 1024
#define SMEM_FLOATS (OFF_OUT + 16 * C_)    // 34048 floats = 136192 bytes

__global__ __launch_bounds__(256)
void m2f_attn_kernel(const float* __restrict__ x, const float* __restrict__ z,
                     const float* __restrict__ Wq, const float* __restrict__ bq,
                     float* __restrict__ ws) {
  extern __shared__ float smem[];
  float* Ksm   = smem + OFF_K;
  float* qsm   = smem + OFF_Q;
  float* attn  = smem + OFF_ATTN;
  float* outsm = smem + OFF_OUT;

  const int tid = threadIdx.x;
  const int blk = blockIdx.x;
  const int b = blk >> 3;
  const int h = blk & 7;

  // ---- Phase 0: stream K block (392x64 f32, contiguous in x) into LDS ----
  // CDNA5 async copy: global_load_async_to_lds_b128 (tracked by ASYNCcnt),
  // per-lane: LDS dest offset in VDST vgpr, 64-bit global address in vgpr pair.
  const float* ksrc = x + (size_t)b * (C_ * HW_) + (size_t)h * (KPH * C_);
  for (int i = tid; i < (KPH * C_) / 4; i += 256) {
    unsigned lds_off =
        (unsigned)(size_t)(__attribute__((address_space(3))) char*)(Ksm + i * 4);
    unsigned long long ga = (unsigned long long)(size_t)(ksrc + i * 4);
    asm volatile("global_load_async_to_lds_b128 %0, %1, off"
                 :: "v"(lds_off), "v"(ga) : "memory");
  }
  // zero pad rows 392..399 and the output accumulator
  for (int i = KPH * C_ + tid; i < KPH_PAD * C_; i += 256) Ksm[i] = 0.0f;
  for (int i = tid; i < 16 * C_; i += 256) outsm[i] = 0.0f;

  // ---- Phase 0b (overlapped with async copies): q tile (16x64), scale folded in ----
  const float scale = 0.07216878364870322f; // 192^-0.5
  for (int idx = tid; idx < 16 * C_; idx += 256) {
    const int t = idx >> 6;
    const int c2 = idx & 63;
    float val = 0.0f;
    if (t < M_) {
      const int tg = h * M_ + t;
      const int m  = tg >> 3;              // q row in (6,512) view
      const int i  = ((tg & 7) << 6) + c2; // q col in (6,512) view
      float acc = bq[i];
      const float* zr = z  + ((size_t)b * M_ + m) * DIM_;
      const float* wr = Wq + (size_t)i * DIM_;
      #pragma unroll 4
      for (int d = 0; d < DIM_; ++d) acc += zr[d] * wr[d];
      val = acc * scale;
    }
    qsm[idx] = val;
  }

#if defined(HAVE_WAIT_ASYNC)
  __builtin_amdgcn_s_wait_asynccnt(0);
#else
  asm volatile("s_wait_asynccnt 0" ::: "memory");
#endif
  __syncthreads();

  const int wave = tid >> 5;
  const int lane = tid & 31;
  const int half = lane >> 4;   // 0: K/M-low pair, 1: high pair (WMMA layout)
  const int lrow = lane & 15;

  // ---- Phase 1: logits tiles = q(16x64) @ K^T via v_wmma_f32_16x16x4_f32 ----
  for (int nt = wave; nt < NTILES; nt += 8) {
    v8f acc = {0, 0, 0, 0, 0, 0, 0, 0};
    const float* qa0 = qsm + lrow * C_;
    const float* kb0 = Ksm + (nt * 16 + lrow) * C_;
    #pragma unroll
    for (int k = 0; k < 16; ++k) {
      const int kk = 4 * k + 2 * half;
      v2f a;  a.x = qa0[kk];  a.y = qa0[kk + 1];   // A[M=lrow][K=kk..kk+1]
      v2f bb; bb.x = kb0[kk]; bb.y = kb0[kk + 1];  // B[K=kk..kk+1][N=lrow]
      acc = __builtin_amdgcn_wmma_f32_16x16x4_f32(false, a, false, bb,
                                                  (short)0, acc, false, false);
    }
    const int key = nt * 16 + lrow;   // C/D layout: col = lane%16
    const bool valid = key < KPH;
    #pragma unroll
    for (int r = 0; r < 8; ++r) {     // row = r + 8*(lane/16)
      attn[(r + 8 * half) * KPH_PAD + key] = valid ? acc[r] : -3.0e38f;
    }
  }
  __syncthreads();

  // ---- Phase 2: softmax per query row (wave w handles row w); waves 6-7 zero pad rows ----
  if (wave < M_) {
    float* rowp = attn + wave * KPH_PAD;
    float mx = -3.0e38f;
    for (int c = lane; c < KPH; c += 32) mx = fmaxf(mx, rowp[c]);
    #pragma unroll
    for (int off = 16; off > 0; off >>= 1) mx = fmaxf(mx, __shfl_xor(mx, off, 32));
    float sum = 0.0f;
    for (int c = lane; c < KPH_PAD; c += 32) {
      const float e = __expf(rowp[c] - mx);  // pad cols -> exp(-huge) = 0
      rowp[c] = e;
      sum += e;
    }
    #pragma unroll
    for (int off = 16; off > 0; off >>= 1) sum += __shfl_xor(sum, off, 32);
    const float inv = 1.0f / sum;
    for (int c = lane; c < KPH_PAD; c += 32) rowp[c] *= inv;
  } else if (wave >= 6) {
    const int t0 = tid - 192;
    for (int p = t0; p < (16 - M_) * KPH_PAD; p += 64) attn[M_ * KPH_PAD + p] = 0.0f;
  }
  __syncthreads();

  // ---- Phase 3: out = attn(16x400) @ K(400x64) via WMMA, cross-wave reduce in LDS ----
  v8f oacc[4] = {};
  for (int nt = wave; nt < NTILES; nt += 8) {
    #pragma unroll
    for (int j = 0; j < 4; ++j) {
      const int kk = 4 * j + 2 * half;
      v2f a;
      a.x = attn[lrow * KPH_PAD + nt * 16 + kk];
      a.y = attn[lrow * KPH_PAD + nt * 16 + kk + 1];
      const float* kr0 = Ksm + (nt * 16 + kk) * C_;
      const float* kr1 = kr0 + C_;
      #pragma unroll
      for (int cb = 0; cb < 4; ++cb) {
        v2f bb; bb.x = kr0[cb * 16 + lrow]; bb.y = kr1[cb * 16 + lrow];
        oacc[cb] = __builtin_amdgcn_wmma_f32_16x16x4_f32(false, a, false, bb,
                                                         (short)0, oacc[cb], false, false);
      }
    }
  }
  #pragma unroll
  for (int cb = 0; cb < 4; ++cb) {
    #pragma unroll
    for (int r = 0; r < 8; ++r) {
      atomicAdd(&outsm[(r + 8 * half) * C_ + cb * 16 + lrow], oacc[cb][r]);
    }
  }
  __syncthreads();

  // ---- Phase 4: write this head's slice of out2 (b, m, h*64 + c) to workspace ----
  for (int idx = tid; idx < M_ * C_; idx += 256) {
    const int m2 = idx >> 6;
    const int c2 = idx & 63;
    ws[((size_t)b * M_ + m2) * INNER_ + h * C_ + c2] = outsm[m2 * C_ + c2];
  }
}

// ---- Kernel 2: result = z + out2 @ Wo^T + bo ----
__global__ __launch_bounds__(192)
void m2f_proj_kernel(const float* __restrict__ z, const float* __restrict__ Wo,
                     const float* __restrict__ bo, const float* __restrict__ ws,
                     float* __restrict__ out) {
  __shared__ float row[INNER_];
  const int blk = blockIdx.x;   // b*6 + m2
  const int tid = threadIdx.x;
  for (int i = tid; i < INNER_; i += 192) row[i] = ws[(size_t)blk * INNER_ + i];
  __syncthreads();
  const int d = tid;            // 0..191
  float acc = bo[d] + z[(size_t)blk * DIM_ + d];
  const float* wr = Wo + (size_t)d * INNER_;
  #pragma unroll 4
  for (int i = 0; i < INNER_; i += 4) {
    acc += row[i] * wr[i] + row[i + 1] * wr[i + 1] +
           row[i + 2] * wr[i + 2] + row[i + 3] * wr[i + 3];
  }
  out[(size_t)blk * DIM_ + d] = acc;
}

extern "C" void kernel_launch(void* const* d_in, const int* in_sizes, int n_in,
                              void* d_out, int out_size, void* d_ws, size_t ws_size,
                              hipStream_t stream) {
  (void)in_sizes; (void)n_in; (void)out_size; (void)ws_size;
  const float* x  = (const float*)d_in[0];
  const float* z  = (const float*)d_in[1];
  const float* Wq = (const float*)d_in[2];
  const float* bq = (const float*)d_in[3];
  const float* Wo = (const float*)d_in[4];
  const float* bo = (const float*)d_in[5];
  float* ws  = (float*)d_ws;   // (32, 6, 512) f32 = 384 KB
  float* out = (float*)d_out;  // (32, 6, 192) f32

  const size_t smem = SMEM_FLOATS * sizeof(float);  // ~136 KB of CDNA5's 320 KB LDS
  m2f_attn_kernel<<<B_ * HEADS, 256, smem, stream>>>(x, z, Wq, bq, ws);
  m2f_proj_kernel<<<B_ * M_, 192, 0, stream>>>(z, Wo, bo, ws, out);
}